// HistogramEqualizer_54528904790401
// MI455X (gfx1250) — compile-verified
//
#include <hip/hip_runtime.h>
#include <hip/hip_bf16.h>
#include <stdint.h>

// ---------------------------------------------------------------------------
// HistogramEqualizer for MI455X (gfx1250).
// Memory-bound: ~0.85 GB of traffic => ~37us at 23.3 TB/s. No matrix ops, so
// the CDNA5 features that matter are async global->LDS copies (ASYNCcnt,
// double-buffered with partial s_wait_asynccnt), LDS ds_add atomics for the
// histogram, and NT cache hints to keep the 201MB batch hot in the 192MB L2.
// ---------------------------------------------------------------------------

#define BIMG 16
#define HW   (1024 * 1024)
#define NB   512
#define TILE 2048   // pixels per channel staged in LDS per iteration
#define BDIM 256    // block size for the streaming kernels

// async issues per wave per tile: (TILE*4 bytes) / (BDIM*16) rounds * 3 channels
#define ISSUES_PER_TILE (((TILE * 4) / (BDIM * 16)) * 3)   // = 6

typedef float          vf4  __attribute__((ext_vector_type(4)));
typedef unsigned char  vuc4 __attribute__((ext_vector_type(4)));

__device__ __forceinline__ bool finitef(float v) {
  return (__float_as_uint(v) & 0x7F800000u) != 0x7F800000u;
}
__device__ __forceinline__ unsigned fenc(float f) {   // order-preserving float->uint
  unsigned u = __float_as_uint(f);
  return (u & 0x80000000u) ? ~u : (u | 0x80000000u);
}
__device__ __forceinline__ float fdec(unsigned u) {
  return __uint_as_float((u & 0x80000000u) ? (u ^ 0x80000000u) : ~u);
}
__device__ __forceinline__ float nanf_() { return __uint_as_float(0x7FC00000u); }

// ---- CDNA5 async global->LDS copy (tracked by ASYNCcnt) -------------------
__device__ __forceinline__ void async_g2l_b128(const void* g, void* l) {
  unsigned           lds = (unsigned)(unsigned long long)l;  // low 32b of generic = LDS offset
  unsigned long long ga  = (unsigned long long)g;
  asm volatile("global_load_async_to_lds_b128 %0, %1, off"
               :: "v"(lds), "v"(ga) : "memory");
}
template <int N>
__device__ __forceinline__ void wait_async() {
  asm volatile("s_wait_asynccnt %0" :: "i"(N) : "memory");
}

// ---------------------------------------------------------------------------
// Pass 0: init workspace (hist=0, min=+inf, max=-inf, sums/cnts=0)
// ---------------------------------------------------------------------------
__global__ void k_init(unsigned* minEnc, unsigned* maxEnc, unsigned* hist,
                       float* sums, unsigned* cnts) {
  int i = blockIdx.x * blockDim.x + threadIdx.x;
  if (i < BIMG) { minEnc[i] = 0xFF800000u; maxEnc[i] = 0x007FFFFFu; }
  if (i < BIMG * NB) hist[i] = 0u;
  if (i < BIMG * 3) { sums[i] = 0.0f; cnts[i] = 0u; }
}

// ---------------------------------------------------------------------------
// Pass 1: masked min/max over all 3 channels (mask = valid & isfinite(ch2))
// ---------------------------------------------------------------------------
__device__ __forceinline__ void mm1(unsigned char mk, float a0, float a1, float a2,
                                    float& lo, float& hi) {
  if (mk && finitef(a2)) {
    lo = fminf(lo, fminf(a0, fminf(a1, a2)));
    hi = fmaxf(hi, fmaxf(a0, fmaxf(a1, a2)));
  }
}

__global__ void k_minmax(const float* __restrict__ x, const unsigned char* __restrict__ mask,
                         unsigned* minEnc, unsigned* maxEnc) {
  int b = blockIdx.y;
  const int chunk = HW / gridDim.x;
  const int base  = blockIdx.x * chunk;
  const vf4*  x0 = (const vf4*)(x + (size_t)(b * 3 + 0) * HW);
  const vf4*  x1 = (const vf4*)(x + (size_t)(b * 3 + 1) * HW);
  const vf4*  x2 = (const vf4*)(x + (size_t)(b * 3 + 2) * HW);
  const vuc4* m4 = (const vuc4*)(mask + (size_t)b * HW);

  float lo = __uint_as_float(0x7F800000u);   // +inf
  float hi = __uint_as_float(0xFF800000u);   // -inf
  for (int q = base / 4 + threadIdx.x; q < (base + chunk) / 4; q += BDIM) {
    vuc4 mm = m4[q];
    vf4  a2 = x2[q];
    if (!(mm.x | mm.y | mm.z | mm.w)) continue;
    vf4 a0 = x0[q];
    vf4 a1 = x1[q];
    mm1(mm.x, a0.x, a1.x, a2.x, lo, hi);
    mm1(mm.y, a0.y, a1.y, a2.y, lo, hi);
    mm1(mm.z, a0.z, a1.z, a2.z, lo, hi);
    mm1(mm.w, a0.w, a1.w, a2.w, lo, hi);
  }
  __shared__ float slo[BDIM], shi[BDIM];
  slo[threadIdx.x] = lo; shi[threadIdx.x] = hi;
  __syncthreads();
  for (int off = BDIM / 2; off > 0; off >>= 1) {
    if (threadIdx.x < off) {
      slo[threadIdx.x] = fminf(slo[threadIdx.x], slo[threadIdx.x + off]);
      shi[threadIdx.x] = fmaxf(shi[threadIdx.x], shi[threadIdx.x + off]);
    }
    __syncthreads();
  }
  if (threadIdx.x == 0) {
    atomicMin(&minEnc[b], fenc(slo[0]));
    atomicMax(&maxEnc[b], fenc(shi[0]));
  }
}

// ---------------------------------------------------------------------------
// Pass 2: masked 512-bin histogram. Channels staged global->LDS with async
// b128 copies, DOUBLE-BUFFERED: tile k+1 is issued, then s_wait_asynccnt 6
// (async loads complete in order => exactly tile k has landed) overlaps the
// DMA with ds_add_u32 binning. LDS: 2x24KB tiles + 2KB hist = 50KB.
// ---------------------------------------------------------------------------
__device__ __forceinline__ void binadd(float v, float xmin, float rng, unsigned* lh) {
  float t = (v - xmin) / rng;                      // matches reference formula
  int i = (int)floorf(t * (float)NB);
  i = i < 0 ? 0 : (i > NB - 1 ? NB - 1 : i);
  atomicAdd(&lh[i], 1u);
}

__device__ __forceinline__ void issue_tile(const float* x, int b, int t0, float* buf) {
  #pragma unroll
  for (int c = 0; c < 3; ++c) {
    const char* gsrc = (const char*)(x + (size_t)(b * 3 + c) * HW + t0);
    char*       ldst = (char*)&buf[c * TILE];
    #pragma unroll
    for (int off = 0; off < TILE * 4; off += BDIM * 16)
      async_g2l_b128(gsrc + off + threadIdx.x * 16, ldst + off + threadIdx.x * 16);
  }
}

__global__ void k_hist(const float* __restrict__ x, const unsigned char* __restrict__ mask,
                       const unsigned* minEnc, const unsigned* maxEnc, unsigned* hist) {
  __shared__ __align__(16) float tiles[2][3 * TILE];   // 2 x 24 KB staging
  __shared__ unsigned lh[NB];                          // 2 KB LDS histogram

  int b = blockIdx.y;
  float xmin = fdec(minEnc[b]);
  float rng  = fdec(maxEnc[b]) - xmin;

  for (int i = threadIdx.x; i < NB; i += BDIM) lh[i] = 0u;

  const int chunk  = HW / gridDim.x;
  const int base   = blockIdx.x * chunk;
  const int ntiles = chunk / TILE;
  unsigned* gh = hist + (size_t)b * NB;

  issue_tile(x, b, base, tiles[0]);                    // prefetch tile 0

  for (int k = 0; k < ntiles; ++k) {
    if (k + 1 < ntiles) {
      issue_tile(x, b, base + (k + 1) * TILE, tiles[(k + 1) & 1]);
      wait_async<ISSUES_PER_TILE>();   // tile k landed; tile k+1 still in flight
    } else {
      wait_async<0>();
    }
    __syncthreads();                   // all waves' tile-k copies visible (+ lh zeroed)

    const float* buf = tiles[k & 1];
    const vuc4*  m4  = (const vuc4*)(mask + (size_t)b * HW + base + k * TILE);
    for (int q = threadIdx.x; q < TILE / 4; q += BDIM) {
      vuc4 mm = m4[q];
      vf4 a0 = ((const vf4*)buf)[q];
      vf4 a1 = ((const vf4*)(buf + TILE))[q];
      vf4 a2 = ((const vf4*)(buf + 2 * TILE))[q];
      if (mm.x && finitef(a2.x)) { binadd(a0.x, xmin, rng, lh); binadd(a1.x, xmin, rng, lh); binadd(a2.x, xmin, rng, lh); }
      if (mm.y && finitef(a2.y)) { binadd(a0.y, xmin, rng, lh); binadd(a1.y, xmin, rng, lh); binadd(a2.y, xmin, rng, lh); }
      if (mm.z && finitef(a2.z)) { binadd(a0.z, xmin, rng, lh); binadd(a1.z, xmin, rng, lh); binadd(a2.z, xmin, rng, lh); }
      if (mm.w && finitef(a2.w)) { binadd(a0.w, xmin, rng, lh); binadd(a1.w, xmin, rng, lh); binadd(a2.w, xmin, rng, lh); }
    }
    __syncthreads();                   // reads of tiles[k&1] done before it is re-issued
  }

  for (int i = threadIdx.x; i < NB; i += BDIM) {
    unsigned v = lh[i];
    if (v) atomicAdd(&gh[i], v);
  }
}

// ---------------------------------------------------------------------------
// Pass 3: per-image CDF + piecewise-linear (m,b) tables. Counts are integer-
// valued floats < 2^24, so the tree scan is bit-exact vs jnp.cumsum.
// ---------------------------------------------------------------------------
__global__ void k_cdf(const unsigned* __restrict__ hist,
                      const unsigned* minEnc, const unsigned* maxEnc,
                      float* __restrict__ M, float* __restrict__ Bb) {
  int b = blockIdx.x;
  int t = threadIdx.x;  // 512 threads
  __shared__ float s[NB];
  __shared__ float cn[NB];
  s[t] = (float)hist[(size_t)b * NB + t];
  __syncthreads();
  for (int off = 1; off < NB; off <<= 1) {
    float v   = s[t];
    float add = (t >= off) ? s[t - off] : 0.0f;
    __syncthreads();
    s[t] = v + add;
    __syncthreads();
  }
  float total = s[NB - 1];
  cn[t] = s[t] / total;
  __syncthreads();
  if (t < NB - 1) {
    float xmin = fdec(minEnc[b]);
    float step = (fdec(maxEnc[b]) - xmin) / (float)NB;
    float ci   = xmin + step * ((float)t + 0.5f);
    float cip  = xmin + step * ((float)t + 1.5f);
    float m    = (cn[t + 1] - cn[t]) / (cip - ci);
    M [(size_t)b * NB + t] = m;
    Bb[(size_t)b * NB + t] = cn[t] - m * ci;
  }
}

// ---------------------------------------------------------------------------
// Pass 4: remap every pixel. NT loads (last read of batch) + NT stores
// (output never re-read) so the batch stays hot in L2. Also accumulates
// per-(b,c) masked sums/counts for the nanmean "invalid" fallback.
// ---------------------------------------------------------------------------
__global__ void k_map(const float* __restrict__ x, const unsigned char* __restrict__ mask,
                      const unsigned* minEnc, const unsigned* maxEnc,
                      const float* __restrict__ M, const float* __restrict__ Bb,
                      float* __restrict__ out, float* sums, unsigned* cnts) {
  __shared__ float sm[NB], sb[NB];
  __shared__ float red[BDIM];
  int b = blockIdx.y;
  for (int i = threadIdx.x; i < NB; i += BDIM) {
    sm[i] = M[(size_t)b * NB + i];
    sb[i] = Bb[(size_t)b * NB + i];
  }
  __syncthreads();

  float xmin = fdec(minEnc[b]);
  float step = (fdec(maxEnc[b]) - xmin) / (float)NB;

  const int chunk = HW / gridDim.x;
  const int base  = blockIdx.x * chunk;
  const vuc4* m4  = (const vuc4*)(mask + (size_t)b * HW);

  float sAcc[3] = {0.f, 0.f, 0.f};
  float cAcc[3] = {0.f, 0.f, 0.f};

  for (int q = base / 4 + threadIdx.x; q < (base + chunk) / 4; q += BDIM) {
    vuc4 mm = m4[q];
    unsigned char mks[4] = {mm.x, mm.y, mm.z, mm.w};
    #pragma unroll
    for (int c = 0; c < 3; ++c) {
      const vf4* xc = (const vf4*)(x + (size_t)(b * 3 + c) * HW);
      vf4*       oc = (vf4*)(out + (size_t)(b * 3 + c) * HW);
      vf4 v = __builtin_nontemporal_load(&xc[q]);
      vf4 o;
      #pragma unroll
      for (int j = 0; j < 4; ++j) {
        float vj = v[j];
        float r;
        if (mks[j] && finitef(vj)) {
          float u = (vj - xmin) / step - 0.5f;       // == searchsorted(centers,x,'right')-1
          int i = (int)floorf(u);
          i = i < 0 ? 0 : (i > NB - 2 ? NB - 2 : i);
          r = (sm[i] * vj + sb[i]) * 2.0f - 1.0f;    // *(TMAX-TMIN)+TMIN
          sAcc[c] += r;
          cAcc[c] += 1.0f;
        } else {
          r = nanf_();
        }
        o[j] = r;
      }
      __builtin_nontemporal_store(o, &oc[q]);
    }
  }

  // block-reduce the 6 accumulators, one global atomic each
  #pragma unroll
  for (int k = 0; k < 6; ++k) {
    __syncthreads();
    red[threadIdx.x] = (k < 3) ? sAcc[k] : cAcc[k - 3];
    __syncthreads();
    for (int off = BDIM / 2; off > 0; off >>= 1) {
      if (threadIdx.x < off) red[threadIdx.x] += red[threadIdx.x + off];
      __syncthreads();
    }
    if (threadIdx.x == 0) {
      if (k < 3) atomicAdd(&sums[b * 3 + k], red[0]);
      else       atomicAdd(&cnts[b * 3 + (k - 3)], (unsigned)red[0]);
    }
  }
}

// ---------------------------------------------------------------------------
// Pass 5: invalid[b][c] = nanmean fallback values
// ---------------------------------------------------------------------------
__global__ void k_invalid(const float* sums, const unsigned* cnts, float* inval) {
  __shared__ float v[BIMG * 3];
  __shared__ float fallback;
  int t = threadIdx.x;
  if (t < BIMG * 3) v[t] = sums[t] / (float)cnts[t];   // cnt==0 -> 0/0 = NaN
  __syncthreads();
  if (t == 0) {
    float s = 0.f; int n = 0;
    for (int i = 0; i < BIMG * 3; ++i)
      if (!(v[i] != v[i])) { s += v[i]; ++n; }          // nanmean semantics
    fallback = n ? s / (float)n : nanf_();
  }
  __syncthreads();
  if (t < BIMG * 3) inval[t] = finitef(v[t]) ? v[t] : fallback;
}

// ---------------------------------------------------------------------------
// Pass 6: fill NaN (invalid) pixels. Early-exits whole blocks when the
// channel was fully valid (the case for this input) -> near-zero cost.
// ---------------------------------------------------------------------------
__global__ void k_fill(const unsigned* cnts, const float* inval, float* out) {
  int b = blockIdx.y;
  if (cnts[b * 3 + 0] == HW && cnts[b * 3 + 1] == HW && cnts[b * 3 + 2] == HW) return;
  const int chunk = HW / gridDim.x;
  const int base  = blockIdx.x * chunk;
  #pragma unroll
  for (int c = 0; c < 3; ++c) {
    float iv   = inval[b * 3 + c];
    float* oc  = out + (size_t)(b * 3 + c) * HW;
    for (int p = base + threadIdx.x; p < base + chunk; p += BDIM) {
      float v = oc[p];
      if (v != v) oc[p] = iv;
    }
  }
}

// ---------------------------------------------------------------------------
extern "C" void kernel_launch(void* const* d_in, const int* in_sizes, int n_in,
                              void* d_out, int out_size, void* d_ws, size_t ws_size,
                              hipStream_t stream) {
  (void)in_sizes; (void)n_in; (void)out_size; (void)ws_size;
  const float*         batch = (const float*)d_in[0];
  const unsigned char* mask  = (const unsigned char*)d_in[1];
  float*               out   = (float*)d_out;

  char* ws = (char*)d_ws;
  unsigned* minEnc = (unsigned*)(ws + 0);          // 16 u32
  unsigned* maxEnc = (unsigned*)(ws + 64);         // 16 u32
  unsigned* hist   = (unsigned*)(ws + 128);        // 16*512 u32 (32 KB)
  float*    M      = (float*)(ws + 32896);         // 16*512 f32
  float*    Bb     = (float*)(ws + 65664);         // 16*512 f32
  float*    sums   = (float*)(ws + 98432);         // 48 f32
  unsigned* cnts   = (unsigned*)(ws + 98624);      // 48 u32
  float*    inval  = (float*)(ws + 98816);         // 48 f32

  k_init   <<<dim3(32),        dim3(BDIM), 0, stream>>>(minEnc, maxEnc, hist, sums, cnts);
  k_minmax <<<dim3(64,  BIMG), dim3(BDIM), 0, stream>>>(batch, mask, minEnc, maxEnc);
  k_hist   <<<dim3(128, BIMG), dim3(BDIM), 0, stream>>>(batch, mask, minEnc, maxEnc, hist);
  k_cdf    <<<dim3(BIMG),      dim3(NB),   0, stream>>>(hist, minEnc, maxEnc, M, Bb);
  k_map    <<<dim3(128, BIMG), dim3(BDIM), 0, stream>>>(batch, mask, minEnc, maxEnc, M, Bb, out, sums, cnts);
  k_invalid<<<dim3(1),         dim3(64),   0, stream>>>(sums, cnts, inval);
  k_fill   <<<dim3(64,  BIMG), dim3(BDIM), 0, stream>>>(cnts, inval, out);
}